// SparseGraphGatedAttentionLayer_90220083020121
// MI455X (gfx1250) — compile-verified
//
#include <hip/hip_runtime.h>

#define IN_F   256
#define OUT_F  128
#define LDSTR  132   // LDS row stride in dwords (132 % 64 == 4 -> 16 lanes hit 16 distinct banks)

typedef __attribute__((ext_vector_type(2))) float v2f;
typedef __attribute__((ext_vector_type(8))) float v8f;

// ---------------------------------------------------------------------------
// Zero the accumulator (d_out) — float4 stores, grid-stride.
// ---------------------------------------------------------------------------
__global__ void zero_kernel(float4* __restrict__ p, long n4) {
    long i = (long)blockIdx.x * blockDim.x + threadIdx.x;
    long stride = (long)gridDim.x * blockDim.x;
    float4 z = make_float4(0.f, 0.f, 0.f, 0.f);
    for (; i < n4; i += stride) p[i] = z;
}

// ---------------------------------------------------------------------------
// Fused GEMMs: h = x@W  then (via LDS band)  h1 = h@a1, h2 = h@a2.
// Block = 256 threads = 8 waves; block handles one 16-row band of nodes;
// wave w owns output columns [16w, 16w+16).
// ---------------------------------------------------------------------------
__global__ __launch_bounds__(256)
void gemm_fused_kernel(const float* __restrict__ x, const float* __restrict__ W,
                       const float* __restrict__ A1, const float* __restrict__ A2,
                       float* __restrict__ h, float* __restrict__ h1,
                       float* __restrict__ h2, int n_nodes) {
    __shared__ float lh[16 * LDSTR];

    const int lane = threadIdx.x & 31;
    const int wave = threadIdx.x >> 5;   // 0..7
    const int n0   = wave * 16;          // N-tile base in OUT_F
    const int half = lane >> 4;          // 0: K+{0,1}, 1: K+{2,3}
    const int l16  = lane & 15;
    const int row0 = blockIdx.x * 16;
    const bool full = (row0 + 16 <= n_nodes);   // uniform across block

    // clamp load row so EXEC stays all-ones through the WMMA loop
    int mrow = row0 + l16;
    int arow = (mrow < n_nodes) ? mrow : (n_nodes - 1);

    // ---- GEMM 1: 16x16 tile of h, K = 256 in steps of 4 -------------------
    v8f c = {};
    const float* xr = x + (long)arow * IN_F + half * 2;
    for (int k = 0; k < IN_F; k += 4) {
        v2f a;
        a.x = xr[k];
        a.y = xr[k + 1];
        int kb = k + half * 2;
        v2f b;
        b.x = W[(long)kb * OUT_F + n0 + l16];
        b.y = W[(long)(kb + 1) * OUT_F + n0 + l16];
        c = __builtin_amdgcn_wmma_f32_16x16x4_f32(false, a, false, b,
                                                  (short)0, c, false, false);
    }

    // scatter tile: VGPR i -> row (i + 8*half), col (n0 + l16)
    {
        float* hp = h + (long)(row0 + 8 * half) * OUT_F + n0 + l16;
        float* lp = lh + (8 * half) * LDSTR + n0 + l16;
        if (full) {
            #pragma unroll
            for (int i = 0; i < 8; ++i) {
                lp[i * LDSTR] = c[i];
                hp[(long)i * OUT_F] = c[i];
            }
        } else {
            #pragma unroll
            for (int i = 0; i < 8; ++i) {
                lp[i * LDSTR] = c[i];
                if (row0 + i + 8 * half < n_nodes) hp[(long)i * OUT_F] = c[i];
            }
        }
    }
    __syncthreads();

    // ---- GEMM 2: h1/h2 tiles from the LDS band, K = 128 -------------------
    v8f c1 = {}, c2 = {};
    for (int k = 0; k < OUT_F; k += 4) {
        int ka = k + half * 2;
        v2f a = *(const v2f*)&lh[l16 * LDSTR + ka];   // 8B-aligned (ka even)
        v2f b1, b2;
        b1.x = A1[(long)ka * OUT_F + n0 + l16];
        b1.y = A1[(long)(ka + 1) * OUT_F + n0 + l16];
        b2.x = A2[(long)ka * OUT_F + n0 + l16];
        b2.y = A2[(long)(ka + 1) * OUT_F + n0 + l16];
        c1 = __builtin_amdgcn_wmma_f32_16x16x4_f32(false, a, false, b1,
                                                   (short)0, c1, false, false);
        c2 = __builtin_amdgcn_wmma_f32_16x16x4_f32(false, a, false, b2,
                                                   (short)0, c2, false, false);
    }
    {
        long base = (long)(row0 + 8 * half) * OUT_F + n0 + l16;
        float* p1 = h1 + base;
        float* p2 = h2 + base;
        if (full) {
            #pragma unroll
            for (int i = 0; i < 8; ++i) {
                p1[(long)i * OUT_F] = c1[i];
                p2[(long)i * OUT_F] = c2[i];
            }
        } else {
            #pragma unroll
            for (int i = 0; i < 8; ++i) {
                if (row0 + i + 8 * half < n_nodes) {
                    p1[(long)i * OUT_F] = c1[i];
                    p2[(long)i * OUT_F] = c2[i];
                }
            }
        }
    }
}

// ---------------------------------------------------------------------------
// Edge phase: one wave per edge, 4 feats per lane (float4 gathers from L2-
// resident h/h1/h2), sigmoid gate, scatter-add into acc[src].
// ---------------------------------------------------------------------------
__global__ __launch_bounds__(256)
void edge_kernel(const long long* __restrict__ ei, const float* __restrict__ ev,
                 const float* __restrict__ h, const float* __restrict__ h1,
                 const float* __restrict__ h2, float* __restrict__ acc,
                 int n_edges) {
    int edge = (int)(((long)blockIdx.x * blockDim.x + threadIdx.x) >> 5);
    if (edge >= n_edges) return;                 // uniform per wave
    int lane = threadIdx.x & 31;

    int src = (int)ei[edge];
    int dst = (int)ei[(long)n_edges + edge];
    float e = ev[edge];

    long fo = (long)lane * 4;
    float4 g1 = *(const float4*)(h1 + (long)src * OUT_F + fo);
    float4 g2 = *(const float4*)(h2 + (long)dst * OUT_F + fo);
    float4 hd = *(const float4*)(h  + (long)dst * OUT_F + fo);

    float4 m;
    m.x = e * hd.x / (1.f + __expf(-(g1.x + g2.x)));
    m.y = e * hd.y / (1.f + __expf(-(g1.y + g2.y)));
    m.z = e * hd.z / (1.f + __expf(-(g1.z + g2.z)));
    m.w = e * hd.w / (1.f + __expf(-(g1.w + g2.w)));

    float* out = acc + (long)src * OUT_F + fo;
    atomicAdd(out + 0, m.x);
    atomicAdd(out + 1, m.y);
    atomicAdd(out + 2, m.z);
    atomicAdd(out + 3, m.w);
}

// ---------------------------------------------------------------------------
// ELU in place on d_out.
// ---------------------------------------------------------------------------
__global__ void elu_kernel(float* __restrict__ p, long n) {
    long i = (long)blockIdx.x * blockDim.x + threadIdx.x;
    long stride = (long)gridDim.x * blockDim.x;
    for (; i < n; i += stride) {
        float v = p[i];
        p[i] = (v > 0.f) ? v : (__expf(v) - 1.f);
    }
}

// ---------------------------------------------------------------------------
extern "C" void kernel_launch(void* const* d_in, const int* in_sizes, int n_in,
                              void* d_out, int out_size, void* d_ws, size_t ws_size,
                              hipStream_t stream) {
    const float*     x  = (const float*)d_in[0];
    const long long* ei = (const long long*)d_in[1];   // int64 [2, M]
    const float*     ev = (const float*)d_in[2];
    const float*     W  = (const float*)d_in[3];
    const float*     A1 = (const float*)d_in[4];
    const float*     A2 = (const float*)d_in[5];

    const int n_nodes = in_sizes[0] / IN_F;
    const int n_edges = in_sizes[2];
    const long nh = (long)n_nodes * OUT_F;

    float* h   = (float*)d_ws;
    float* h1  = h  + nh;
    float* h2  = h1 + nh;
    float* acc = (float*)d_out;

    // 1) zero accumulator
    zero_kernel<<<(int)((nh / 4 + 255) / 256), 256, 0, stream>>>((float4*)acc, nh / 4);

    // 2) fused GEMMs (one block per 16-row band)
    int bands = (n_nodes + 15) / 16;
    gemm_fused_kernel<<<bands, 256, 0, stream>>>(x, W, A1, A2, h, h1, h2, n_nodes);

    // 3) edge scatter (8 edges per 256-thread block)
    int eb = (n_edges + 7) / 8;
    edge_kernel<<<eb, 256, 0, stream>>>(ei, ev, h, h1, h2, acc, n_edges);

    // 4) ELU epilogue in place
    elu_kernel<<<(int)((nh + 255) / 256), 256, 0, stream>>>(acc, nh);
}